// GDLoss_206158430835
// MI455X (gfx1250) — compile-verified
//
#include <hip/hip_runtime.h>
#include <math.h>

// ext_vector types (plain clang vectors: required for nontemporal + WMMA builtins)
typedef float v2f __attribute__((ext_vector_type(2)));
typedef float v4f __attribute__((ext_vector_type(4)));
typedef float v8f __attribute__((ext_vector_type(8)));

#define DEG2RAD 0.017453292519943295f

// Great-circle central angle (atan2 formulation), all math in f32 like the reference.
__device__ __forceinline__ float great_circle_pair(float pl, float pg, float tl, float tg) {
    // rescale: col0 -> lat degrees (x*25+15, pred clamped), col1 -> lng degrees (x*360)
    float lat1 = fminf(fmaxf(pl * 25.0f + 15.0f, -90.0f), 90.0f) * DEG2RAD;
    float lng1 = pg * (360.0f * DEG2RAD);
    float lat2 = (tl * 25.0f + 15.0f) * DEG2RAD;
    float lng2 = tg * (360.0f * DEG2RAD);

    float s1, c1, s2, c2, sd, cd;
    __sincosf(lat1, &s1, &c1);
    __sincosf(lat2, &s2, &c2);
    __sincosf(lng2 - lng1, &sd, &cd);

    float t1  = c2 * sd;
    float t2  = c1 * s2 - s1 * c2 * cd;
    float num = sqrtf(t1 * t1 + t2 * t2);
    float den = s1 * s2 + c1 * c2 * cd;
    return atan2f(num, den);
}

// Wave32 sum-reduction via V_WMMA_F32_16X16X4_F32.
// A(16x4): lane m (0-15) supplies K0 (=partial) and K1 (=0); lane m+16 supplies K2,K3.
// B(4x16) = all ones  =>  D[m][n] = partial[m] + partial[m+16] for every n.
// Sum of a lane's 8 C VGPRs = sum of 8 rows; lane0 covers rows 0-7, lane16 rows 8-15.
__device__ __forceinline__ float wave_reduce_wmma(float x) {
    v2f a; a.x = x;    a.y = 0.0f;
    v2f b; b.x = 1.0f; b.y = 1.0f;
    v8f c = {};
    c = __builtin_amdgcn_wmma_f32_16x16x4_f32(
        /*neg_a=*/false, a, /*neg_b=*/false, b,
        /*c_mod=*/(short)0, c, /*reuse_a=*/false, /*reuse_b=*/false);
    float s = c[0] + c[1] + c[2] + c[3] + c[4] + c[5] + c[6] + c[7];
    return __shfl(s, 0, 32) + __shfl(s, 16, 32);
}

__global__ void __launch_bounds__(256)
gd_partial_kernel(const v4f* __restrict__ pred, const v4f* __restrict__ tru,
                  float* __restrict__ partials, int nquads) {
    int tid    = blockIdx.x * blockDim.x + threadIdx.x;
    int stride = gridDim.x * blockDim.x;

    float acc = 0.0f;
    // Stream once, bypass near caches (256 MB working set > 192 MB L2): NT hint.
    for (int i = tid; i < nquads; i += stride) {
        v4f p = __builtin_nontemporal_load(&pred[i]); // (lat0,lng0,lat1,lng1)
        v4f t = __builtin_nontemporal_load(&tru[i]);
        acc += great_circle_pair(p.x, p.y, t.x, t.y);
        acc += great_circle_pair(p.z, p.w, t.z, t.w);
    }

    // EXEC is all-1s here (no divergence outside the counted loop) -> WMMA legal.
    float wave_sum = wave_reduce_wmma(acc);

    __shared__ float lds[8];
    int lane = threadIdx.x & 31;
    int wave = threadIdx.x >> 5;
    if (lane == 0) lds[wave] = wave_sum;
    __syncthreads();
    if (threadIdx.x == 0) {
        float bs = 0.0f;
        int nw = blockDim.x >> 5;
        for (int w = 0; w < nw; ++w) bs += lds[w];   // fixed order: deterministic
        partials[blockIdx.x] = bs;
    }
}

__global__ void __launch_bounds__(256)
gd_final_kernel(const float* __restrict__ partials, int n,
                float* __restrict__ out, float inv_b) {
    __shared__ float lds[256];
    float acc = 0.0f;
    for (int i = threadIdx.x; i < n; i += 256) acc += partials[i]; // fixed order
    lds[threadIdx.x] = acc;
    __syncthreads();
    for (int s = 128; s > 0; s >>= 1) {
        if (threadIdx.x < (unsigned)s) lds[threadIdx.x] += lds[threadIdx.x + s];
        __syncthreads();
    }
    if (threadIdx.x == 0) out[0] = lds[0] * inv_b;
}

extern "C" void kernel_launch(void* const* d_in, const int* in_sizes, int n_in,
                              void* d_out, int out_size, void* d_ws, size_t ws_size,
                              hipStream_t stream) {
    const v4f* pred = (const v4f*)d_in[0];
    const v4f* tru  = (const v4f*)d_in[1];
    float* out      = (float*)d_out;
    float* partials = (float*)d_ws;          // NB floats of scratch

    const long total  = (long)in_sizes[0];   // B*2 floats (pred is (B,2) f32)
    const int  nquads = (int)(total / 4);    // one v4f = 2 coordinate pairs
    const int  NB = 4096, T = 256;           // 1M threads / 32K waves: saturates HBM
    const float inv_b = (float)(2.0 / (double)total);  // 1/B

    gd_partial_kernel<<<NB, T, 0, stream>>>(pred, tru, partials, nquads);
    gd_final_kernel<<<1, T, 0, stream>>>(partials, NB, out, inv_b);
}